// ICLoss_22857815949971
// MI455X (gfx1250) — compile-verified
//
#include <hip/hip_runtime.h>

typedef float v2f __attribute__((ext_vector_type(2)));
typedef float v8f __attribute__((ext_vector_type(8)));

#define STAT_STRIDE 8   // {cnt, sp, sy, spp, syy, spy, pad, pad}
#define FULL_MASK   0xffffffffu

// ---------------------------------------------------------------------------
// Exact 32-lane f32 sum on the CDNA5 matrix pipe.
// V_WMMA_F32_16X16X4_F32: D[m][n] = sum_k A[m][k]*B[k][n] (+C).
// With B == 1 and A = {v, 0} per lane, every per-lane value appears exactly
// once in A, so each column of D sums to partial totals; adding the 8 C VGPRs
// and folding lane halves with SWAPX16 yields the full wave sum in all lanes.
// Must be called with EXEC all-ones (wave-uniform control flow).
// ---------------------------------------------------------------------------
__device__ __forceinline__ float wave_sum32(float v) {
    v2f a; a[0] = v;    a[1] = 0.0f;
    v2f b; b[0] = 1.0f; b[1] = 1.0f;
    v8f c = {};
    c = __builtin_amdgcn_wmma_f32_16x16x4_f32(false, a, false, b, (short)0, c,
                                              false, false);
    float r = ((c[0] + c[1]) + (c[2] + c[3])) + ((c[4] + c[5]) + (c[6] + c[7]));
    // rows 8..15 of D live in lanes 16..31: fold halves (ds_swizzle SWAPX16)
    r += __int_as_float(__builtin_amdgcn_ds_swizzle(__float_as_int(r), 0x401f));
    return r;
}

__device__ __forceinline__ void flush_seg(int segid, float* stats, unsigned lane,
                                          float& c, float& sp, float& sy,
                                          float& pp, float& yy, float& py) {
    if (segid >= 0) {
        float tc  = wave_sum32(c);
        float tsp = wave_sum32(sp);
        float tsy = wave_sum32(sy);
        float tpp = wave_sum32(pp);
        float tyy = wave_sum32(yy);
        float tpy = wave_sum32(py);
        if (lane == 0) {
            float* st = stats + (size_t)segid * STAT_STRIDE;
            atomicAdd(st + 0, tc);
            atomicAdd(st + 1, tsp);
            atomicAdd(st + 2, tsy);
            atomicAdd(st + 3, tpp);
            atomicAdd(st + 4, tyy);
            atomicAdd(st + 5, tpy);
        }
    }
    c = sp = sy = pp = yy = py = 0.0f;
}

// ---------------------------------------------------------------------------
// Zero the per-segment stat table (n_segments lives on device).
// ---------------------------------------------------------------------------
__global__ __launch_bounds__(256) void ic_zero(float* __restrict__ stats,
                                               const int* __restrict__ nsegp) {
    const int n = (*nsegp) * STAT_STRIDE;
    for (int i = blockIdx.x * blockDim.x + threadIdx.x; i < n;
         i += gridDim.x * blockDim.x)
        stats[i] = 0.0f;
}

// ---------------------------------------------------------------------------
// Streaming pass: each wave owns a contiguous 128-aligned span. Fast path
// keeps 6 per-lane f32 accumulators; segment changes flush via WMMA + atomics.
// ---------------------------------------------------------------------------
__global__ __launch_bounds__(256)
void ic_accumulate(const float* __restrict__ pred, const float* __restrict__ yv,
                   const int* __restrict__ seg, float* __restrict__ stats,
                   int N) {
    const unsigned lane   = threadIdx.x & 31u;
    const unsigned nwaves = (gridDim.x * blockDim.x) >> 5;
    const unsigned wave   = blockIdx.x * (blockDim.x >> 5) + (threadIdx.x >> 5);

    unsigned span = ((unsigned)N + nwaves - 1u) / nwaves;
    span = (span + 127u) & ~127u;                 // keep float4 alignment
    const unsigned start = wave * span;
    if (start >= (unsigned)N) return;             // wave-uniform exit
    const unsigned end   = (start + span < (unsigned)N) ? start + span : (unsigned)N;
    const unsigned niter = (end - start + 127u) >> 7;

    int   cur = -1;
    float ac = 0.f, asp = 0.f, asy = 0.f, app = 0.f, ayy = 0.f, apy = 0.f;

    for (unsigned it = 0; it < niter; ++it) {
        const unsigned i0 = start + it * 128u + lane * 4u;
        float p0, p1, p2, p3, q0, q1, q2, q3;
        int   s0, s1, s2, s3, nin;
        if (i0 + 3u < end) {                      // bulk: 16B vector loads
            const float4 pv = *(const float4*)(pred + i0);
            const float4 qv = *(const float4*)(yv + i0);
            const int4   sv = *(const int4*)(seg + i0);
            p0 = pv.x; p1 = pv.y; p2 = pv.z; p3 = pv.w;
            q0 = qv.x; q1 = qv.y; q2 = qv.z; q3 = qv.w;
            s0 = sv.x; s1 = sv.y; s2 = sv.z; s3 = sv.w;
            nin = 4;
        } else {                                  // guarded tail
            p0 = p1 = p2 = p3 = 0.f; q0 = q1 = q2 = q3 = 0.f;
            s0 = s1 = s2 = s3 = -2; nin = 0;
            if (i0 + 0u < end) { p0 = pred[i0 + 0]; q0 = yv[i0 + 0]; s0 = seg[i0 + 0]; ++nin; }
            if (i0 + 1u < end) { p1 = pred[i0 + 1]; q1 = yv[i0 + 1]; s1 = seg[i0 + 1]; ++nin; }
            if (i0 + 2u < end) { p2 = pred[i0 + 2]; q2 = yv[i0 + 2]; s2 = seg[i0 + 2]; ++nin; }
            if (i0 + 3u < end) { p3 = pred[i0 + 3]; q3 = yv[i0 + 3]; s3 = seg[i0 + 3]; ++nin; }
        }

        auto accum = [&]() {
            ac  += (float)nin;
            asp += (p0 + p1) + (p2 + p3);
            asy += (q0 + q1) + (q2 + q3);
            app  = fmaf(p0, p0, fmaf(p1, p1, fmaf(p2, p2, fmaf(p3, p3, app))));
            ayy  = fmaf(q0, q0, fmaf(q1, q1, fmaf(q2, q2, fmaf(q3, q3, ayy))));
            apy  = fmaf(p0, q0, fmaf(p1, q1, fmaf(p2, q2, fmaf(p3, q3, apy))));
        };

        const bool same = (s0 == cur) & (s1 == cur) & (s2 == cur) & (s3 == cur);
        if (__builtin_amdgcn_ballot_w32(same) == FULL_MASK) {
            accum();                               // hot path: segment run continues
        } else {
            flush_seg(cur, stats, lane, ac, asp, asy, app, ayy, apy);
            const int u = __builtin_amdgcn_readfirstlane(s0);
            const bool ok = (s0 == u || s0 < 0) && (s1 == u || s1 < 0) &&
                            (s2 == u || s2 < 0) && (s3 == u || s3 < 0);
            if (__builtin_amdgcn_ballot_w32(ok) == FULL_MASK) {
                cur = u;                           // new uniform run (ids sorted)
                accum();
            } else {
                cur = -1;                          // rare: boundary inside tile
                const int   ss[4] = {s0, s1, s2, s3};
                const float pp4[4] = {p0, p1, p2, p3};
                const float qq4[4] = {q0, q1, q2, q3};
#pragma unroll
                for (int j = 0; j < 4; ++j) {
                    if (ss[j] >= 0) {
                        float* st = stats + (size_t)ss[j] * STAT_STRIDE;
                        atomicAdd(st + 0, 1.0f);
                        atomicAdd(st + 1, pp4[j]);
                        atomicAdd(st + 2, qq4[j]);
                        atomicAdd(st + 3, pp4[j] * pp4[j]);
                        atomicAdd(st + 4, qq4[j] * qq4[j]);
                        atomicAdd(st + 5, pp4[j] * qq4[j]);
                    }
                }
            }
        }
    }
    flush_seg(cur, stats, lane, ac, asp, asy, app, ayy, apy);
}

// ---------------------------------------------------------------------------
// Per-day IC + masked mean (single block; D ~ 2000).
// ---------------------------------------------------------------------------
__global__ __launch_bounds__(256)
void ic_finalize(const float* __restrict__ stats, const int* __restrict__ skipp,
                 const int* __restrict__ nsegp, float* __restrict__ out) {
    const int   D    = *nsegp;
    const float skip = (float)(*skipp);
    float ic_sum = 0.f, nv = 0.f;
    for (int d = (int)threadIdx.x; d < D; d += (int)blockDim.x) {
        const float* st = stats + (size_t)d * STAT_STRIDE;
        const float cnt = st[0], sp = st[1], sy = st[2];
        const float spp = st[3], syy = st[4], spy = st[5];
        const float cs = fmaxf(cnt, 2.0f);
        const float mp = sp / cs, my = sy / cs;
        const float varp = fmaxf(spp - cnt * mp * mp, 0.f) / (cs - 1.f);
        const float vary = fmaxf(syy - cnt * my * my, 0.f) / (cs - 1.f);
        const float sdp = sqrtf(varp), sdy = sqrtf(vary);
        const float denom = fmaxf(cs * sdp * sdy, 1e-30f);
        const float ic = (spy - cnt * mp * my) / denom;
        const bool valid = (cnt >= skip) && (sdp >= 1e-12f) && (sdy >= 1e-12f);
        ic_sum += valid ? ic : 0.f;
        nv     += valid ? 1.f : 0.f;
    }
    ic_sum = wave_sum32(ic_sum);                  // WMMA wave reduction
    nv     = wave_sum32(nv);
    __shared__ float sh[16];
    const unsigned w = threadIdx.x >> 5, lane = threadIdx.x & 31u;
    if (lane == 0) { sh[w] = ic_sum; sh[w + 8] = nv; }
    __syncthreads();
    if (threadIdx.x == 0) {
        float a = 0.f, b = 0.f;
        for (int i = 0; i < 8; ++i) { a += sh[i]; b += sh[i + 8]; }
        out[0] = -(a / b);
    }
}

extern "C" void kernel_launch(void* const* d_in, const int* in_sizes, int n_in,
                              void* d_out, int out_size, void* d_ws, size_t ws_size,
                              hipStream_t stream) {
    (void)n_in; (void)out_size; (void)ws_size;
    const float* pred = (const float*)d_in[0];
    const float* yv   = (const float*)d_in[1];
    const int*   seg  = (const int*)d_in[2];
    const int*   skip = (const int*)d_in[3];
    const int*   nseg = (const int*)d_in[4];
    float* stats = (float*)d_ws;                  // 8 * n_segments floats
    const int N = in_sizes[0];

    ic_zero<<<64, 256, 0, stream>>>(stats, nseg);

    int waves  = (N + 2047) / 2048;               // ~2K contiguous elems / wave
    int blocks = (waves + 7) / 8;                 // 8 wave32 per 256-thread block
    if (blocks < 1) blocks = 1;
    if (blocks > 8192) blocks = 8192;
    ic_accumulate<<<blocks, 256, 0, stream>>>(pred, yv, seg, stats, N);

    ic_finalize<<<1, 256, 0, stream>>>(stats, skip, nseg, (float*)d_out);
}